// SparseCoder_50319836840655
// MI455X (gfx1250) — compile-verified
//
#include <hip/hip_runtime.h>
#include <math.h>

typedef __attribute__((ext_vector_type(2))) float v2f;
typedef __attribute__((ext_vector_type(8))) float v8f;

#define N_ATOMS 256
#define CHAN    128
#define BATCH   8
#define HW      1024
#define M_TOTAL (BATCH * HW)   // 8192 rows
#define MAX_ITER 20
#define LMBDA   0.1f

// ---------------- setup kernels ----------------

// PhiPair[kp*CHAN + c] = (Phi[2kp][c], Phi[2kp+1][c])   kp in [0,128), c in [0,128)
__global__ __launch_bounds__(256) void phi_pair_kernel(
    const float* __restrict__ Phi, float2* __restrict__ PhiPair) {
  int idx = blockIdx.x * blockDim.x + threadIdx.x;   // over 128*128
  if (idx < (N_ATOMS / 2) * CHAN) {
    int kp = idx / CHAN;
    int c  = idx % CHAN;
    float2 v;
    v.x = Phi[(2 * kp    ) * CHAN + c];
    v.y = Phi[(2 * kp + 1) * CHAN + c];
    PhiPair[idx] = v;
  }
}

// PhiTPair[kp*N_ATOMS + c] = (Phi[c][2kp], Phi[c][2kp+1])  kp in [0,64), c in [0,256)
__global__ __launch_bounds__(256) void phit_pair_kernel(
    const float* __restrict__ Phi, float2* __restrict__ PhiTPair) {
  int idx = blockIdx.x * blockDim.x + threadIdx.x;   // over 64*256
  if (idx < (CHAN / 2) * N_ATOMS) {
    int kp = idx / N_ATOMS;
    int c  = idx % N_ATOMS;
    PhiTPair[idx] = *(const float2*)(Phi + (size_t)c * CHAN + 2 * kp);
  }
}

// L = || Phi^T Phi ||_F  (G is 128x128, each entry a 256-dot; one block)
__global__ __launch_bounds__(256) void compute_L_kernel(
    const float* __restrict__ Phi, float* __restrict__ Lout) {
  __shared__ float red[256];
  int tid = threadIdx.x;
  float sumsq = 0.f;
  for (int idx = tid; idx < CHAN * CHAN; idx += 256) {
    int i = idx / CHAN, j = idx % CHAN;
    float dot = 0.f;
    for (int a = 0; a < N_ATOMS; ++a)
      dot += Phi[a * CHAN + i] * Phi[a * CHAN + j];
    sumsq += dot * dot;
  }
  red[tid] = sumsq;
  __syncthreads();
  for (int s = 128; s > 0; s >>= 1) {
    if (tid < s) red[tid] += red[tid + s];
    __syncthreads();
  }
  if (tid == 0) Lout[0] = sqrtf(red[0]);
}

// x [B,C,hw] -> x_flat [B*hw, C]
__global__ __launch_bounds__(256) void x_transpose_kernel(
    const float* __restrict__ x, float* __restrict__ xf) {
  int o = blockIdx.x * blockDim.x + threadIdx.x;     // over 8192*128, [m][c]
  if (o < M_TOTAL * CHAN) {
    int c = o & (CHAN - 1);
    int m = o >> 7;              // b*1024 + p
    int p = m & (HW - 1);
    int b = m >> 10;
    xf[o] = x[(size_t)(b * CHAN + c) * HW + p];
  }
}

__global__ __launch_bounds__(256) void zero_kernel(float* __restrict__ ptr, int n) {
  int i = blockIdx.x * blockDim.x + threadIdx.x;
  if (i < n) ptr[i] = 0.f;
}

// ---------------- FISTA iteration kernels ----------------
// K1: R[8192x128] = Y[8192x256] @ Phi[256x128] - Xf
// 256 thr = 8 waves; wave -> 16-row M tile; paired Phi in 128KB dynamic LDS.
// 2-stage software pipeline: prefetch next A/B fragments under current WMMAs.
__global__ __launch_bounds__(256) void residual_kernel(
    const float* __restrict__ Y, const float2* __restrict__ PhiPair,
    const float* __restrict__ Xf, float* __restrict__ R) {
  extern __shared__ float lds1[];              // 128*128 float2 = 128 KB
  {
    const float4* src = (const float4*)PhiPair;
    float4* dst = (float4*)lds1;
    for (int i = threadIdx.x; i < (N_ATOMS * CHAN) / 4; i += 256) dst[i] = src[i];
  }
  __syncthreads();
  const v2f* sB = (const v2f*)lds1;            // [kp][col]

  int wave = threadIdx.x >> 5;
  int lane = threadIdx.x & 31;
  int half = lane >> 4;                         // K-pair select for A/B frags
  int lm   = lane & 15;
  int mBase = (blockIdx.x * 8 + wave) * 16;
  const float* yrow = Y + (size_t)(mBase + lm) * N_ATOMS;

  v8f acc[8] = {};

  // pipeline prologue (k = 0)
  v2f a_cur = *(const v2f*)(yrow + 2 * half);
  v2f b_cur[8];
#pragma unroll
  for (int n = 0; n < 8; ++n) b_cur[n] = sB[half * CHAN + n * 16 + lm];

#pragma unroll 4
  for (int k = 0; k < N_ATOMS - 4; k += 4) {
    int kpn = ((k + 4) >> 1) + half;
    v2f a_nxt = *(const v2f*)(yrow + (k + 4) + 2 * half);
    v2f b_nxt[8];
#pragma unroll
    for (int n = 0; n < 8; ++n) b_nxt[n] = sB[kpn * CHAN + n * 16 + lm];
#pragma unroll
    for (int n = 0; n < 8; ++n)
      acc[n] = __builtin_amdgcn_wmma_f32_16x16x4_f32(
          false, a_cur, false, b_cur[n], (short)0, acc[n], false, false);
    a_cur = a_nxt;
#pragma unroll
    for (int n = 0; n < 8; ++n) b_cur[n] = b_nxt[n];
  }
#pragma unroll
  for (int n = 0; n < 8; ++n)
    acc[n] = __builtin_amdgcn_wmma_f32_16x16x4_f32(
        false, a_cur, false, b_cur[n], (short)0, acc[n], false, false);

#pragma unroll
  for (int n = 0; n < 8; ++n) {
    int col = n * 16 + lm;
#pragma unroll
    for (int i = 0; i < 8; ++i) {
      int row = mBase + half * 8 + i;           // C layout: half0 M=0..7, half1 M=8..15
      size_t idx = (size_t)row * CHAN + col;
      R[idx] = acc[n][i] - Xf[idx];
    }
  }
}

// K2: G = Y - (R @ PhiT)/L; A_new = shrink(G, lambda/L); Y = A_new + mom*(A_new - A_old)
// In-place update of A and Y (row-local, hazard-free).
__global__ __launch_bounds__(256) void grad_update_kernel(
    const float* __restrict__ R, const float2* __restrict__ PhiTPair,
    const float* __restrict__ Lptr, float* __restrict__ A,
    float* __restrict__ Yv, float mom) {
  extern __shared__ float lds2[];              // 64*256 float2 = 128 KB
  {
    const float4* src = (const float4*)PhiTPair;
    float4* dst = (float4*)lds2;
    for (int i = threadIdx.x; i < (CHAN * N_ATOMS) / 4; i += 256) dst[i] = src[i];
  }
  __syncthreads();
  const v2f* sB = (const v2f*)lds2;            // [kp][col]

  float invL = 1.0f / Lptr[0];
  float thr  = LMBDA * invL;

  int wave = threadIdx.x >> 5;
  int lane = threadIdx.x & 31;
  int half = lane >> 4;
  int lm   = lane & 15;
  int mBase = (blockIdx.x * 8 + wave) * 16;
  const float* rrow = R + (size_t)(mBase + lm) * CHAN;

  for (int nh = 0; nh < 2; ++nh) {             // two N-passes of 128 atoms each
    int nb = nh * 128;                          // atom base (in floats)
    v8f acc[8] = {};

    v2f a_cur = *(const v2f*)(rrow + 2 * half);
    v2f b_cur[8];
#pragma unroll
    for (int n = 0; n < 8; ++n)
      b_cur[n] = sB[half * N_ATOMS + nb + n * 16 + lm];

#pragma unroll 4
    for (int k = 0; k < CHAN - 4; k += 4) {
      int kpn = ((k + 4) >> 1) + half;
      v2f a_nxt = *(const v2f*)(rrow + (k + 4) + 2 * half);
      v2f b_nxt[8];
#pragma unroll
      for (int n = 0; n < 8; ++n)
        b_nxt[n] = sB[kpn * N_ATOMS + nb + n * 16 + lm];
#pragma unroll
      for (int n = 0; n < 8; ++n)
        acc[n] = __builtin_amdgcn_wmma_f32_16x16x4_f32(
            false, a_cur, false, b_cur[n], (short)0, acc[n], false, false);
      a_cur = a_nxt;
#pragma unroll
      for (int n = 0; n < 8; ++n) b_cur[n] = b_nxt[n];
    }
#pragma unroll
    for (int n = 0; n < 8; ++n)
      acc[n] = __builtin_amdgcn_wmma_f32_16x16x4_f32(
          false, a_cur, false, b_cur[n], (short)0, acc[n], false, false);

#pragma unroll
    for (int n = 0; n < 8; ++n) {
      int col = nb + n * 16 + lm;
#pragma unroll
      for (int i = 0; i < 8; ++i) {
        int row = mBase + half * 8 + i;
        size_t idx = (size_t)row * N_ATOMS + col;
        float yold = Yv[idx];
        float g = yold - acc[n][i] * invL;
        float mag = fabsf(g) - thr;
        float anew = (mag > 0.f) ? copysignf(mag, g) : 0.f;
        float aold = A[idx];
        A[idx]  = anew;
        Yv[idx] = anew + mom * (anew - aold);
      }
    }
  }
}

// out[b][atom][p] = A[b*1024+p][atom]
__global__ __launch_bounds__(256) void emit_out_kernel(
    const float* __restrict__ A, float* __restrict__ out) {
  int o = blockIdx.x * blockDim.x + threadIdx.x;
  if (o < BATCH * N_ATOMS * HW) {
    int p    = o & (HW - 1);
    int atom = (o >> 10) & (N_ATOMS - 1);
    int b    = o >> 18;
    out[o] = A[(size_t)(b * HW + p) * N_ATOMS + atom];
  }
}

// ---------------- host launcher ----------------
extern "C" void kernel_launch(void* const* d_in, const int* in_sizes, int n_in,
                              void* d_out, int out_size, void* d_ws, size_t ws_size,
                              hipStream_t stream) {
  const float* Phi = (const float*)d_in[0];  // [256,128]
  const float* x   = (const float*)d_in[1];  // [8,128,32,32]
  float* out = (float*)d_out;                // [8,256,32,32]
  float* ws  = (float*)d_ws;

  // workspace layout (floats)
  float2* PhiPair  = (float2*)ws;                       // 16384 float2 = 32768 f
  float2* PhiTPair = (float2*)(ws + 32768);             // 16384 float2 = 32768 f
  float*  Lbuf = ws + 65536;                            // 1 (padded to 32)
  float*  Xf   = ws + 65568;                            // 1,048,576
  float*  A    = Xf + (size_t)M_TOTAL * CHAN;           // 2,097,152
  float*  Yv   = A + (size_t)M_TOTAL * N_ATOMS;         // 2,097,152
  float*  R    = Yv + (size_t)M_TOTAL * N_ATOMS;        // 1,048,576

  // one-time setup
  phi_pair_kernel<<<((N_ATOMS / 2) * CHAN + 255) / 256, 256, 0, stream>>>(Phi, PhiPair);
  phit_pair_kernel<<<((CHAN / 2) * N_ATOMS + 255) / 256, 256, 0, stream>>>(Phi, PhiTPair);
  compute_L_kernel<<<1, 256, 0, stream>>>(Phi, Lbuf);
  x_transpose_kernel<<<(M_TOTAL * CHAN + 255) / 256, 256, 0, stream>>>(x, Xf);
  zero_kernel<<<(2 * M_TOTAL * N_ATOMS + 255) / 256, 256, 0, stream>>>(
      A, 2 * M_TOTAL * N_ATOMS);

  const int mBlocks = M_TOTAL / 128;                         // 64 workgroups
  const size_t ldsBytes = (size_t)N_ATOMS * CHAN * sizeof(float); // 128 KB

  float t = 1.0f;
  for (int it = 0; it < MAX_ITER; ++it) {
    float tn  = 0.5f * (1.0f + sqrtf(1.0f + 4.0f * t * t));
    float mom = (t - 1.0f) / tn;
    residual_kernel<<<mBlocks, 256, ldsBytes, stream>>>(Yv, PhiPair, Xf, R);
    grad_update_kernel<<<mBlocks, 256, ldsBytes, stream>>>(R, PhiTPair, Lbuf, A, Yv, mom);
    t = tn;
  }
  emit_out_kernel<<<(BATCH * N_ATOMS * HW + 255) / 256, 256, 0, stream>>>(A, out);
}